// SEAttention_71614284693685
// MI455X (gfx1250) — compile-verified
//
#include <hip/hip_runtime.h>

typedef __attribute__((ext_vector_type(2))) float v2f;
typedef __attribute__((ext_vector_type(4))) float v4f;
typedef __attribute__((ext_vector_type(8))) float v8f;

#define Bn 32
#define Cn 512
#define Hn 56
#define Wn 56
#define HWn (Hn * Wn)       // 3136
#define CR  32              // C / RED
#define NQUAD ((Hn / 2) * (Wn / 4))  // 28*14 = 392 2x4 patches per plane

// ---------------------------------------------------------------------------
// Kernel 1: per-(b,c) mean over H*W.  Streaming read of x (205 MB), bandwidth
// bound.  One block per plane, float4 (b128) loads, LDS tree reduction.
// ---------------------------------------------------------------------------
__global__ void k_mean(const float* __restrict__ x, float* __restrict__ y) {
    const int plane = blockIdx.x;                    // b*C + c, 0..16383
    const float* p = x + (size_t)plane * HWn;
    const v4f* p4 = (const v4f*)p;
    float s = 0.f;
    for (int i = threadIdx.x; i < HWn / 4; i += 256) {
        v4f v = p4[i];
        s += (v.x + v.y) + (v.z + v.w);
    }
    __shared__ float red[256];
    red[threadIdx.x] = s;
    __syncthreads();
    for (int off = 128; off > 0; off >>= 1) {
        if (threadIdx.x < off) red[threadIdx.x] += red[threadIdx.x + off];
        __syncthreads();
    }
    if (threadIdx.x == 0) y[plane] = red[0] * (1.0f / HWn);
}

// ---------------------------------------------------------------------------
// Kernel 2a: Y1 = relu(Y @ W1^T),  Y[32,512], W1[32,512] -> Y1[32,32].
// D = A x B with A = Y, B[k,n] = W1[n,k].  4 output tiles of 16x16, one wave
// per tile, K-loop of 128 x V_WMMA_F32_16X16X4_F32 (exact fp32 matrix path).
// ---------------------------------------------------------------------------
__global__ void k_fc1(const float* __restrict__ y, const float* __restrict__ W1,
                      float* __restrict__ y1) {
    const int wave = threadIdx.x >> 5;       // 0..3
    const int lane = threadIdx.x & 31;
    const int Mt = (wave >> 1) * 16;
    const int Nt = (wave & 1) * 16;
    const int mn = lane & 15;                // row of A frag / col of B frag
    const int kh = (lane >> 4) * 2;          // K sub-offset per ISA A16x4 layout
    const float* arow = y  + (size_t)(Mt + mn) * Cn;
    const float* brow = W1 + (size_t)(Nt + mn) * Cn;   // B = W1^T -> row-major W1
    v8f acc = {};
    for (int k0 = 0; k0 < Cn; k0 += 4) {
        v2f a = *(const v2f*)(arow + k0 + kh);
        v2f b = *(const v2f*)(brow + k0 + kh);
        acc = __builtin_amdgcn_wmma_f32_16x16x4_f32(
            false, a, false, b, (short)0, acc, false, false);
    }
#pragma unroll
    for (int r = 0; r < 8; ++r) {
        int row = Mt + r + (lane >> 4) * 8;  // C/D layout: VGPR r -> M=r / r+8
        int col = Nt + mn;
        float v = acc[r];
        y1[row * CR + col] = v > 0.f ? v : 0.f;
    }
}

// ---------------------------------------------------------------------------
// Kernel 2b: gate = sigmoid(Y1 @ W2^T),  Y1[32,32], W2[512,32] -> gate[32,512].
// 64 tiles of 16x16, K=32 -> 8 WMMA steps.  16 blocks x 4 waves.
// ---------------------------------------------------------------------------
__global__ void k_fc2(const float* __restrict__ y1, const float* __restrict__ W2,
                      float* __restrict__ gate) {
    const int wave = threadIdx.x >> 5;
    const int lane = threadIdx.x & 31;
    const int tile = blockIdx.x * 4 + wave;  // 0..63
    const int Mt = (tile & 1) * 16;          // M in {0,16}
    const int Nt = (tile >> 1) * 16;         // N in {0,...,496}
    const int mn = lane & 15;
    const int kh = (lane >> 4) * 2;
    const float* arow = y1 + (size_t)(Mt + mn) * CR;
    const float* brow = W2 + (size_t)(Nt + mn) * CR;   // B = W2^T -> row-major W2
    v8f acc = {};
#pragma unroll
    for (int k0 = 0; k0 < CR; k0 += 4) {
        v2f a = *(const v2f*)(arow + k0 + kh);
        v2f b = *(const v2f*)(brow + k0 + kh);
        acc = __builtin_amdgcn_wmma_f32_16x16x4_f32(
            false, a, false, b, (short)0, acc, false, false);
    }
#pragma unroll
    for (int r = 0; r < 8; ++r) {
        int row = Mt + r + (lane >> 4) * 8;
        int col = Nt + mn;
        float v = acc[r];
        gate[row * Cn + col] = 1.0f / (1.0f + __expf(-v));
    }
}

// ---------------------------------------------------------------------------
// Kernel 3: fused gate + windowed-mean scaling.
// out = xr * pooled  where pooled = g * mean2x2(x)  and  xr = g * x,
// so out_i = x_i * (g^2 * mean2x2(x)).  One block per (b,c) plane; each
// thread owns a 2x4 patch (two adjacent 2x2 windows) via two b128 loads and
// two b128 stores (16B-aligned: plane base 12544 B, row 224 B, col 16 B).
// ---------------------------------------------------------------------------
__global__ void k_scale(const float* __restrict__ x, const float* __restrict__ gate,
                        float* __restrict__ out) {
    const int plane = blockIdx.x;            // b*C + c
    const float g = gate[plane];
    const float g2q = g * g * 0.25f;
    const float* px = x + (size_t)plane * HWn;
    float* po = out + (size_t)plane * HWn;
    for (int t = threadIdx.x; t < NQUAD; t += 256) {
        int wh = t / (Wn / 4);               // 0..27  (window-row)
        int wq = t - wh * (Wn / 4);          // 0..13  (float4 column group)
        int base0 = (2 * wh) * Wn + 4 * wq;
        int base1 = base0 + Wn;
        v4f x0 = *(const v4f*)(px + base0);  // row 2*wh,   cols 4wq..4wq+3
        v4f x1 = *(const v4f*)(px + base1);  // row 2*wh+1, cols 4wq..4wq+3
        float sA = ((x0.x + x0.y) + (x1.x + x1.y)) * g2q;  // left 2x2 window
        float sB = ((x0.z + x0.w) + (x1.z + x1.w)) * g2q;  // right 2x2 window
        v4f o0, o1;
        o0.x = x0.x * sA; o0.y = x0.y * sA; o0.z = x0.z * sB; o0.w = x0.w * sB;
        o1.x = x1.x * sA; o1.y = x1.y * sA; o1.z = x1.z * sB; o1.w = x1.w * sB;
        *(v4f*)(po + base0) = o0;
        *(v4f*)(po + base1) = o1;
    }
}

// ---------------------------------------------------------------------------
extern "C" void kernel_launch(void* const* d_in, const int* in_sizes, int n_in,
                              void* d_out, int out_size, void* d_ws, size_t ws_size,
                              hipStream_t stream) {
    const float* x  = (const float*)d_in[0];   // [32,512,56,56]
    const float* W1 = (const float*)d_in[1];   // [32,512]
    const float* W2 = (const float*)d_in[2];   // [512,32]
    float* out = (float*)d_out;

    float* y    = (float*)d_ws;                // [32,512]  channel means
    float* y1   = y + Bn * Cn;                 // [32,32]   relu(fc1)
    float* gate = y1 + Bn * CR;                // [32,512]  sigmoid(fc2)

    k_mean <<<Bn * Cn, 256, 0, stream>>>(x, y);
    k_fc1  <<<1,       128, 0, stream>>>(y, W1, y1);
    k_fc2  <<<16,      128, 0, stream>>>(y1, W2, gate);
    k_scale<<<Bn * Cn, 256, 0, stream>>>(x, gate, out);
}